// LigerFusedLinearCrossEntropyHelion_37271726194737
// MI455X (gfx1250) — compile-verified
//
#include <hip/hip_runtime.h>
#include <hip/hip_bf16.h>
#include <cstddef>

// ---------------------------------------------------------------------------
// Fused linear cross-entropy fwd + grad_x for MI455X (gfx1250, wave32).
// HBM-bound (~2.1 GB traffic @ 23.3 TB/s >> fp32 WMMA compute), so both GEMMs
// use V_WMMA_F32_16X16X4_F32 at full fp32 fidelity. Tiles are moved with
// GLOBAL_LOAD_ASYNC_TO_LDS_B128 (ASYNCcnt) into double-buffered LDS so DMA of
// tile i+1 overlaps the WMMAs of tile i.
// ---------------------------------------------------------------------------

typedef float v2f __attribute__((ext_vector_type(2)));
typedef float v8f __attribute__((ext_vector_type(8)));
typedef int   v4i __attribute__((ext_vector_type(4)));

constexpr int BT = 4096;
constexpr int H  = 2048;
constexpr int V  = 32000;

constexpr int BM  = 128;    // tile rows (M)
constexpr int BN  = 128;    // tile cols (N)
constexpr int BK  = 32;     // K depth per stage
constexpr int LDK = BK + 4; // 36: even pad -> 8B-aligned K-pairs, bank spread

// ---- CDNA5 async global->LDS path (fallback: sync copy), compile-safe ------
#if defined(__AMDGCN__) && __has_builtin(__builtin_amdgcn_global_load_async_to_lds_b128)
#define ASYNC_COPY 1
#else
#define ASYNC_COPY 0
#endif

__device__ __forceinline__ void copy_b128(const float* g, float* l) {
#if ASYNC_COPY
  __builtin_amdgcn_global_load_async_to_lds_b128(
      (__attribute__((address_space(1))) v4i*)g,
      (__attribute__((address_space(3))) v4i*)l,
      /*offset=*/0, /*cpol=*/0);
#else
  *reinterpret_cast<float4*>(l) = *reinterpret_cast<const float4*>(g);
#endif
}

#if ASYNC_COPY
#if __has_builtin(__builtin_amdgcn_s_wait_asynccnt)
#define WAIT_ASYNC(n) __builtin_amdgcn_s_wait_asynccnt(n)
#else
#define WAIT_ASYNC(n) asm volatile("s_wait_asynccnt %0" ::"i"(n) : "memory")
#endif
#else
#define WAIT_ASYNC(n) ((void)0)
#endif

__device__ __forceinline__ v8f v8f_zero() {
  v8f z = {0.f, 0.f, 0.f, 0.f, 0.f, 0.f, 0.f, 0.f};
  return z;
}

// ---------------------------------------------------------------------------
// Kernel 1: logits[b,v] = sum_h x[b,h] * w[v,h]       (NT GEMM via WMMA f32)
// ---------------------------------------------------------------------------
__global__ __launch_bounds__(256) void logits_gemm_kernel(
    const float* __restrict__ x, const float* __restrict__ w,
    float* __restrict__ logits) {
  __shared__ float Xs[2][BM * LDK];
  __shared__ float Ws[2][BN * LDK];

  const int tid     = threadIdx.x;
  const int lane    = tid & 31;
  const int wid     = tid >> 5;        // 8 waves
  const int wm      = (wid & 3) * 32;  // 4 waves along M, 32 rows each
  const int wn      = (wid >> 2) * 64; // 2 waves along N, 64 cols each
  const int mBase   = blockIdx.x * BM;
  const int nBase   = blockIdx.y * BN;
  const int halfSel = lane >> 4;       // 0: K=kk..kk+1, 1: K=kk+2..kk+3
  const int l16     = lane & 15;

  // Per-thread fill slots: 128x32 floats per tile = 1024 float4 / 256 thr.
  auto issue = [&](int buf, int kb) {
#pragma unroll
    for (int r = 0; r < 4; ++r) {
      const int i   = tid + r * 256;
      const int row = i >> 3;          // 0..127
      const int kc  = (i & 7) * 4;     // 0..28
      copy_b128(&x[(size_t)(mBase + row) * H + kb + kc],
                &Xs[buf][row * LDK + kc]);
      copy_b128(&w[(size_t)(nBase + row) * H + kb + kc],
                &Ws[buf][row * LDK + kc]);
    }
  };

  v8f acc[2][4];
#pragma unroll
  for (int i = 0; i < 2; ++i)
#pragma unroll
    for (int j = 0; j < 4; ++j) acc[i][j] = v8f_zero();

  constexpr int nK = H / BK;           // 64
  issue(0, 0);                         // prologue: 8 async ops in flight

  for (int ki = 0; ki < nK; ++ki) {
    const int cur = ki & 1;
    if (ki + 1 < nK) {
      issue(cur ^ 1, (ki + 1) * BK);   // DMA next tile into other buffer
      WAIT_ASYNC(8);                   // first batch (cur tile) landed
    } else {
      WAIT_ASYNC(0);
    }
    __syncthreads();                   // all waves' DMA for cur visible

#pragma unroll
    for (int kk = 0; kk < BK; kk += 4) {
      const int ka = kk + halfSel * 2;
      v2f a[2], b[4];
#pragma unroll
      for (int i = 0; i < 2; ++i) {
        const int m = wm + i * 16 + l16;
        a[i][0] = Xs[cur][m * LDK + ka];
        a[i][1] = Xs[cur][m * LDK + ka + 1];
      }
#pragma unroll
      for (int j = 0; j < 4; ++j) {
        const int n = wn + j * 16 + l16;
        b[j][0] = Ws[cur][n * LDK + ka];
        b[j][1] = Ws[cur][n * LDK + ka + 1];
      }
#pragma unroll
      for (int i = 0; i < 2; ++i)
#pragma unroll
        for (int j = 0; j < 4; ++j)
          acc[i][j] = __builtin_amdgcn_wmma_f32_16x16x4_f32(
              false, a[i], false, b[j], (short)0, acc[i][j], false, false);
    }
    __syncthreads();                   // done reading cur before overwrite
  }

  // Epilogue: VGPR r holds M = r + 8*halfSel, N = l16 within each 16x16 tile
#pragma unroll
  for (int i = 0; i < 2; ++i) {
#pragma unroll
    for (int j = 0; j < 4; ++j) {
      const int n  = nBase + wn + j * 16 + l16;
      const int m0 = mBase + wm + i * 16 + halfSel * 8;
#pragma unroll
      for (int r = 0; r < 8; ++r)
        logits[(size_t)(m0 + r) * V + n] = acc[i][j][r];
    }
  }
}

// ---------------------------------------------------------------------------
// Kernel 2: per-row logsumexp, target gather, mean-NLL accumulation
// ---------------------------------------------------------------------------
__global__ __launch_bounds__(256) void softmax_stats_kernel(
    const float* __restrict__ logits, const int* __restrict__ target,
    float* __restrict__ loss, float* __restrict__ lse_out,
    float* __restrict__ neg_out) {
  const int b   = blockIdx.x;
  const int tid = threadIdx.x;
  const float* row = logits + (size_t)b * V;
  __shared__ float red[256];

  float m = -3.4028235e38f;
  for (int v = tid; v < V; v += 256) m = fmaxf(m, row[v]);
  red[tid] = m;
  __syncthreads();
#pragma unroll
  for (int s = 128; s > 0; s >>= 1) {
    if (tid < s) red[tid] = fmaxf(red[tid], red[tid + s]);
    __syncthreads();
  }
  const float rowmax = red[0];
  __syncthreads();

  float s = 0.f;
  for (int v = tid; v < V; v += 256) s += __expf(row[v] - rowmax);
  red[tid] = s;
  __syncthreads();
#pragma unroll
  for (int st = 128; st > 0; st >>= 1) {
    if (tid < st) red[tid] += red[tid + st];
    __syncthreads();
  }

  if (tid == 0) {
    const float lse = rowmax + __logf(red[0]);
    lse_out[b] = lse;
    const float tl = row[target[b]];
    neg_out[b] = -tl;
    atomicAdd(loss, (lse - tl) * (1.0f / (float)BT));
  }
}

// ---------------------------------------------------------------------------
// Kernel 3: grad_x = softmax(logits) @ W
// Raw logits tiles are DMA'd to LDS; exp(raw - lse) is applied at
// fragment-load time (lse is loop-invariant per wave row). TRANS-pipe exps
// co-execute with the WMMA pipe.
// ---------------------------------------------------------------------------
__global__ __launch_bounds__(256) void grad_gemm_kernel(
    const float* __restrict__ logits, const float* __restrict__ lse,
    const float* __restrict__ w, float* __restrict__ gx) {
  __shared__ float Ls[2][BM * LDK];    // raw logits tile [m][k], padded
  __shared__ float Wt[2][BK * BN];     // W tile          [k][n], row-major
  __shared__ float lseS[BM];

  const int tid     = threadIdx.x;
  const int lane    = tid & 31;
  const int wid     = tid >> 5;
  const int wm      = (wid & 3) * 32;
  const int wn      = (wid >> 2) * 64;
  const int mBase   = blockIdx.x * BM;
  const int nBase   = blockIdx.y * BN;
  const int halfSel = lane >> 4;
  const int l16     = lane & 15;

  if (tid < BM) lseS[tid] = lse[mBase + tid];

  auto issue = [&](int buf, int kb) {
#pragma unroll
    for (int r = 0; r < 4; ++r) {
      const int i = tid + r * 256;
      {  // logits tile: 128 rows x 32 k
        const int row = i >> 3;
        const int kc  = (i & 7) * 4;
        copy_b128(&logits[(size_t)(mBase + row) * V + kb + kc],
                  &Ls[buf][row * LDK + kc]);
      }
      {  // W tile: 32 k-rows x 128 n
        const int kr = i >> 5;
        const int nc = (i & 31) * 4;
        copy_b128(&w[(size_t)(kb + kr) * H + nBase + nc],
                  &Wt[buf][kr * BN + nc]);
      }
    }
  };

  v8f acc[2][4];
#pragma unroll
  for (int i = 0; i < 2; ++i)
#pragma unroll
    for (int j = 0; j < 4; ++j) acc[i][j] = v8f_zero();

  constexpr int nK = V / BK;           // 1000
  issue(0, 0);
  __syncthreads();                     // lseS visible (and tile0 in flight)

  // loop-invariant per-wave row lse values for A fragments
  float lseA[2];
#pragma unroll
  for (int i = 0; i < 2; ++i) lseA[i] = lseS[wm + i * 16 + l16];

  for (int ki = 0; ki < nK; ++ki) {
    const int cur = ki & 1;
    if (ki + 1 < nK) {
      issue(cur ^ 1, (ki + 1) * BK);
      WAIT_ASYNC(8);
    } else {
      WAIT_ASYNC(0);
    }
    __syncthreads();

#pragma unroll
    for (int kk = 0; kk < BK; kk += 4) {
      const int ka = kk + halfSel * 2;
      v2f a[2], b[4];
#pragma unroll
      for (int i = 0; i < 2; ++i) {
        const int m = wm + i * 16 + l16;
        a[i][0] = __expf(Ls[cur][m * LDK + ka] - lseA[i]);
        a[i][1] = __expf(Ls[cur][m * LDK + ka + 1] - lseA[i]);
      }
#pragma unroll
      for (int j = 0; j < 4; ++j) {
        const int n = wn + j * 16 + l16;
        b[j][0] = Wt[cur][ka * BN + n];
        b[j][1] = Wt[cur][(ka + 1) * BN + n];
      }
#pragma unroll
      for (int i = 0; i < 2; ++i)
#pragma unroll
        for (int j = 0; j < 4; ++j)
          acc[i][j] = __builtin_amdgcn_wmma_f32_16x16x4_f32(
              false, a[i], false, b[j], (short)0, acc[i][j], false, false);
    }
    __syncthreads();
  }

#pragma unroll
  for (int i = 0; i < 2; ++i) {
#pragma unroll
    for (int j = 0; j < 4; ++j) {
      const int n  = nBase + wn + j * 16 + l16;
      const int m0 = mBase + wm + i * 16 + halfSel * 8;
#pragma unroll
      for (int r = 0; r < 8; ++r)
        gx[(size_t)(m0 + r) * H + n] = acc[i][j][r];
    }
  }
}

// ---------------------------------------------------------------------------
// Launch
// ---------------------------------------------------------------------------
extern "C" void kernel_launch(void* const* d_in, const int* in_sizes, int n_in,
                              void* d_out, int out_size, void* d_ws,
                              size_t ws_size, hipStream_t stream) {
  (void)in_sizes; (void)n_in; (void)out_size; (void)d_ws; (void)ws_size;

  const float* x      = (const float*)d_in[0];
  const float* w      = (const float*)d_in[1];
  const int*   target = (const int*)d_in[2];

  float* out    = (float*)d_out;
  float* loss   = out;                       // [1]
  float* logits = out + 1;                   // [BT*V]
  float* lseo   = logits + (size_t)BT * V;   // [BT]
  float* nego   = lseo + BT;                 // [BT]
  float* gx     = nego + BT;                 // [BT*H]

  (void)hipMemsetAsync(loss, 0, sizeof(float), stream);

  logits_gemm_kernel<<<dim3(BT / BM, V / BN), 256, 0, stream>>>(x, w, logits);
  softmax_stats_kernel<<<dim3(BT), 256, 0, stream>>>(logits, target, loss,
                                                     lseo, nego);
  grad_gemm_kernel<<<dim3(BT / BM, H / BN), 256, 0, stream>>>(logits, lseo, w,
                                                              gx);
}